// MultiRegionRNN_25752623907175
// MI455X (gfx1250) — compile-verified
//
#include <hip/hip_runtime.h>
#include <hip/hip_bf16.h>

// Problem sizes (reference): T=500, B=64, N=512, D=100, O=10, alpha=0.1
#define TT 500
#define BB 64
#define NN 512
#define DD 100
#define OO 10
#define NWG 48            // 48 workgroups x 8 waves = 384 waves = 384 output tiles

typedef float v2f __attribute__((ext_vector_type(2)));
typedef float v8f __attribute__((ext_vector_type(8)));

// ---------------------------------------------------------------------------
// fp32 WMMA GEMM inner loop: accumulates a 16x16 tile of A(16xK) * B(Kx16).
//   A fragment: lane l holds A[l&15][kb], A[l&15][kb+1]   (kb = (l>>4)*2)
//   B fragment: lane l holds B[kb][l&15], B[kb+1][l&15] == W[n0+(l&15)][kb..]
// Both are contiguous float2 loads (8B aligned: k steps by 4, ld*4 % 8 == 0).
// ---------------------------------------------------------------------------
__device__ __forceinline__ v8f wmma_dotk(v8f c, const float* __restrict__ arow,
                                         const float* __restrict__ brow, int K) {
#pragma unroll 8
  for (int k = 0; k < K; k += 4) {
    v2f a = *(const v2f*)(arow + k);
    v2f b = *(const v2f*)(brow + k);
    // (neg_a, A, neg_b, B, c_mod, C, reuse_a, reuse_b)
    c = __builtin_amdgcn_wmma_f32_16x16x4_f32(false, a, false, b, (short)0, c,
                                              false, false);
  }
  return c;
}

// ---------------------------------------------------------------------------
// Monotonic-counter grid barrier (no counter reset -> no reset/arrive race).
// ---------------------------------------------------------------------------
__device__ __forceinline__ void grid_barrier(unsigned* cnt, unsigned* gen,
                                             unsigned iter) {
  __syncthreads();
  if (threadIdx.x == 0) {
    __threadfence();  // release our workgroup's stores to device scope
    unsigned old = __hip_atomic_fetch_add(cnt, 1u, __ATOMIC_ACQ_REL,
                                          __HIP_MEMORY_SCOPE_AGENT);
    if (old == (unsigned)NWG * iter - 1u) {
      __hip_atomic_store(gen, iter, __ATOMIC_RELEASE, __HIP_MEMORY_SCOPE_AGENT);
    } else {
      while ((int)(__hip_atomic_load(gen, __ATOMIC_ACQUIRE,
                                     __HIP_MEMORY_SCOPE_AGENT) - iter) < 0) {
        __builtin_amdgcn_s_sleep(1);
      }
    }
    __threadfence();  // acquire other workgroups' stores
  }
  __syncthreads();
}

// ---------------------------------------------------------------------------
// Workspace zero-init (x state, both r buffers, barrier words).
// ---------------------------------------------------------------------------
__global__ void rnn_init_kernel(float* ws, unsigned* bar) {
  const int total = 3 * BB * NN + 2 * 3 * BB * NN;  // x + r[2]
  for (int i = blockIdx.x * blockDim.x + threadIdx.x; i < total;
       i += gridDim.x * blockDim.x)
    ws[i] = 0.0f;
  if (blockIdx.x == 0 && threadIdx.x < 2) bar[threadIdx.x] = 0u;
}

// ---------------------------------------------------------------------------
// Persistent multi-region RNN kernel. One 16x16 output tile per wave.
//   wave id: region = w>>7 (m1/pmd/s1), bt = (w&127)>>5, nt = w&31
// ---------------------------------------------------------------------------
__global__ __launch_bounds__(256, 1) void rnn_persistent_kernel(
    const float* __restrict__ X,        // [500,64,100]
    const float* __restrict__ Wrec_m1,  // [512,512]
    const float* __restrict__ Wrec_pmd,
    const float* __restrict__ Wrec_s1,
    const float* __restrict__ b_m1,     // [512]
    const float* __restrict__ b_pmd,
    const float* __restrict__ b_s1,
    const float* __restrict__ Wpmd_m1,  // [512,512]
    const float* __restrict__ Ws1_m1,
    const float* __restrict__ Wm1_pmd,
    const float* __restrict__ Win_pmd,  // [512,100]
    const float* __restrict__ Win_s1,   // [512,100]
    const float* __restrict__ W_out,    // [10,512]
    float* __restrict__ out,            // [499,64,10]
    float* __restrict__ ws_x,           // [3,64,512]
    float* __restrict__ ws_r,           // [2,3,64,512]
    unsigned* __restrict__ bar) {
  const int lane   = (int)(threadIdx.x & 31u);
  const int wid    = (int)blockIdx.x * 8 + (int)(threadIdx.x >> 5);
  const int region = wid >> 7;         // 0=m1, 1=pmd, 2=s1
  const int bt     = (wid & 127) >> 5; // batch tile 0..3
  const int nt     = wid & 31;         // n tile 0..31
  const int b0 = bt * 16, n0 = nt * 16;

  const int mrow = lane & 15;          // A row / B col / C col index
  const int khi  = lane >> 4;          // high half-wave flag
  const int kb   = khi * 2;            // per-lane K sub-offset for A/B

  // per-lane fragment base offsets
  const int aoffN = (b0 + mrow) * NN + kb;   // into r buffers (ld=512)
  const int aoffD = (b0 + mrow) * DD + kb;   // into X rows     (ld=100)
  const int woffN = (n0 + mrow) * NN + kb;   // into NxN weights
  const int woffD = (n0 + mrow) * DD + kb;   // into NxD weights

  float* xreg = ws_x + region * (BB * NN);
  const float* bias = (region == 0) ? b_m1 : (region == 1) ? b_pmd : b_s1;
  const float bv = bias[n0 + mrow];

  // out-readout assignment: s1-region blocks (32..34), 640 dots of length 512
  const int ot = (int)(blockIdx.x * 256u + threadIdx.x) - 32 * 256;

  unsigned iter = 0;
  for (int t = 1; t < TT; ++t) {
    const int rb = (t - 1) & 1, wb = t & 1;
    const float* rbuf = ws_r + rb * (3 * BB * NN);
    float* wbuf       = ws_r + wb * (3 * BB * NN) + region * (BB * NN);
    const float* r_m1  = rbuf;
    const float* r_pmd = rbuf + BB * NN;
    const float* r_s1  = rbuf + 2 * BB * NN;
    const float* Xt    = X + t * (BB * DD);

    v8f acc = {0.f, 0.f, 0.f, 0.f, 0.f, 0.f, 0.f, 0.f};
    if (region == 0) {        // m1: r_m1 Wrec^T + r_pmd Wpmd_m1^T + r_s1 Ws1_m1^T
      acc = wmma_dotk(acc, r_m1 + aoffN, Wrec_m1 + woffN, NN);
      acc = wmma_dotk(acc, r_pmd + aoffN, Wpmd_m1 + woffN, NN);
      acc = wmma_dotk(acc, r_s1 + aoffN, Ws1_m1 + woffN, NN);
    } else if (region == 1) { // pmd: r_pmd Wrec^T + r_m1 Wm1_pmd^T + Xt Win_pmd^T
      acc = wmma_dotk(acc, r_pmd + aoffN, Wrec_pmd + woffN, NN);
      acc = wmma_dotk(acc, r_m1 + aoffN, Wm1_pmd + woffN, NN);
      acc = wmma_dotk(acc, Xt + aoffD, Win_pmd + woffD, DD);
    } else {                  // s1: r_s1 Wrec^T + Xt Win_s1^T
      acc = wmma_dotk(acc, r_s1 + aoffN, Wrec_s1 + woffN, NN);
      acc = wmma_dotk(acc, Xt + aoffD, Win_s1 + woffD, DD);
    }

    // leaky update + tanh; C/D layout: VGPR v -> row m = v + khi*8, col n = mrow
#pragma unroll
    for (int v = 0; v < 8; ++v) {
      const int idx = (b0 + v + khi * 8) * NN + n0 + mrow;
      const float xn = 0.9f * xreg[idx] + 0.1f * (acc[v] + bv);
      xreg[idx] = xn;
      wbuf[idx] = tanhf(xn);
    }

    ++iter;
    grid_barrier(bar, bar + 1, iter);

    // readout out[t-1] = r_m1(new) @ W_out^T, overlapped with next step's GEMMs
    if (ot >= 0 && ot < BB * OO) {
      const int b = ot / OO, o = ot - b * OO;
      const float4* r4 = (const float4*)(ws_r + wb * (3 * BB * NN) + b * NN);
      const float4* w4 = (const float4*)(W_out + o * NN);
      float s = 0.f;
#pragma unroll 4
      for (int k = 0; k < NN / 4; ++k) {
        const float4 a = r4[k], c = w4[k];
        s += a.x * c.x + a.y * c.y + a.z * c.z + a.w * c.w;
      }
      out[(t - 1) * (BB * OO) + ot] = s;
    }
  }
}

// ---------------------------------------------------------------------------
extern "C" void kernel_launch(void* const* d_in, const int* in_sizes, int n_in,
                              void* d_out, int out_size, void* d_ws,
                              size_t ws_size, hipStream_t stream) {
  (void)in_sizes; (void)n_in; (void)out_size; (void)ws_size;

  const float* X        = (const float*)d_in[0];
  const float* Wrec_m1  = (const float*)d_in[1];
  const float* Wrec_pmd = (const float*)d_in[2];
  const float* Wrec_s1  = (const float*)d_in[3];
  const float* b_m1     = (const float*)d_in[4];
  const float* b_pmd    = (const float*)d_in[5];
  const float* b_s1     = (const float*)d_in[6];
  const float* Wpmd_m1  = (const float*)d_in[7];
  const float* Ws1_m1   = (const float*)d_in[8];
  const float* Wm1_pmd  = (const float*)d_in[9];
  const float* Win_pmd  = (const float*)d_in[10];
  const float* Win_s1   = (const float*)d_in[11];
  const float* W_out    = (const float*)d_in[12];

  float* wsf       = (float*)d_ws;
  float* ws_x      = wsf;                         // 3*64*512 floats
  float* ws_r      = wsf + 3 * BB * NN;           // 2*3*64*512 floats
  unsigned* bar    = (unsigned*)(wsf + 3 * BB * NN + 2 * 3 * BB * NN);

  rnn_init_kernel<<<256, 256, 0, stream>>>(wsf, bar);
  rnn_persistent_kernel<<<NWG, 256, 0, stream>>>(
      X, Wrec_m1, Wrec_pmd, Wrec_s1, b_m1, b_pmd, b_s1, Wpmd_m1, Ws1_m1,
      Wm1_pmd, Win_pmd, Win_s1, W_out, (float*)d_out, ws_x, ws_r, bar);
}